// PatchExtractor_16939351016313
// MI455X (gfx1250) — compile-verified
//
#include <hip/hip_runtime.h>
#include <math.h>

// MI455X notes:
//  - memory-bound problem: ~1.2 GB total HBM traffic @ 23.3 TB/s => ~52us floor.
//  - wave32; WMMA f32 16x16x4 used for the channel-attention MLP (exact f32,
//    matches reference precision, and exercises the CDNA5 matrix path).
//  - gather kernel orders blocks with b' slow so each phase's 151MB plane set
//    stays L2-resident (192MB L2).

typedef __attribute__((ext_vector_type(2))) float v2f;
typedef __attribute__((ext_vector_type(8))) float v8f;

#define CCH 256
#define HDIM 384
#define HW (384*384)       // 147456
#define NPLANE 1024        // B*C
#define MKPT 4096

// ---------------- Kernel 1: per-plane mean & max over H*W ----------------
__global__ __launch_bounds__(256)
void plane_reduce(const float* __restrict__ seg,
                  float* __restrict__ avg_out, float* __restrict__ max_out) {
    int plane = blockIdx.x;                      // b*C + c
    int tid = threadIdx.x;
    const float4* p4 = (const float4*)(seg + (size_t)plane * HW);
    float s = 0.f;
    float m = -__builtin_inff();
    for (int i = tid; i < HW/4; i += 256) {      // coalesced 512B/wave
        float4 v = p4[i];
        s += v.x + v.y + v.z + v.w;
        m = fmaxf(m, fmaxf(fmaxf(v.x, v.y), fmaxf(v.z, v.w)));
    }
    __shared__ float ls[256];
    __shared__ float lm[256];
    ls[tid] = s; lm[tid] = m;
    __syncthreads();
    for (int off = 128; off > 0; off >>= 1) {
        if (tid < off) {
            ls[tid] += ls[tid + off];
            lm[tid] = fmaxf(lm[tid], lm[tid + off]);
        }
        __syncthreads();
    }
    if (tid == 0) {
        avg_out[plane] = ls[0] * (1.0f / (float)HW);
        max_out[plane] = lm[0];
    }
}

// ---------------- Kernel 2: channel-attention MLP via V_WMMA_F32_16X16X4_F32
// V (16x256): rows 0-3 = avg[b], rows 4-7 = max[b], rows 8-15 = 0.
// h = relu(V @ w1^T); z = h @ w2^T; s[b][c] = sigmoid(z[b][c] + z[b+4][c]).
// 16 waves, wave w owns output columns [16w, 16w+16). K=256 in 64 steps of 4.
// Fragment layout (ISA 7.12.2, 32-bit): A: lane<16 -> M=lane, K={k0,k0+1};
// lane>=16 -> K={k0+2,k0+3}.  B: N=lane&15, same K split.  D: row j+8*(lane>>4).
__global__ __launch_bounds__(512)
void ca_mlp(const float* __restrict__ avg, const float* __restrict__ mx,
            const float* __restrict__ w1, const float* __restrict__ w2,
            float* __restrict__ s_out) {
    __shared__ float V[16][CCH];
    __shared__ float Hd[16][CCH];
    int tid = threadIdx.x;
    for (int i = tid; i < 16*CCH; i += 512) {
        int r = i >> 8, j = i & 255;
        float v = 0.f;
        if (r < 4)      v = avg[r*CCH + j];
        else if (r < 8) v = mx[(r-4)*CCH + j];
        V[r][j] = v;
    }
    __syncthreads();

    int wid  = tid >> 5;
    int lane = tid & 31;
    int half = lane >> 4;
    int l16  = lane & 15;
    int ncol = wid * 16 + l16;                   // output column (N)

    // ---- layer 1 ----
    v8f acc = {0.f,0.f,0.f,0.f,0.f,0.f,0.f,0.f};
    for (int k0 = 0; k0 < CCH; k0 += 4) {
        int kg = k0 + half*2;
        v2f a = { V[l16][kg], V[l16][kg+1] };
        const float* br = w1 + (size_t)ncol*CCH + kg;   // B[K][N] = w1[N][K]
        v2f b = { br[0], br[1] };
        acc = __builtin_amdgcn_wmma_f32_16x16x4_f32(
                  false, a, false, b, (short)0, acc, false, false);
    }
    #pragma unroll
    for (int j = 0; j < 8; ++j)
        Hd[half*8 + j][ncol] = fmaxf(acc[j], 0.f);       // relu; rows 8-15 stay 0
    __syncthreads();

    // ---- layer 2 ----
    v8f acc2 = {0.f,0.f,0.f,0.f,0.f,0.f,0.f,0.f};
    for (int k0 = 0; k0 < CCH; k0 += 4) {
        int kg = k0 + half*2;
        v2f a = { Hd[l16][kg], Hd[l16][kg+1] };
        const float* br = w2 + (size_t)ncol*CCH + kg;
        v2f b = { br[0], br[1] };
        acc2 = __builtin_amdgcn_wmma_f32_16x16x4_f32(
                   false, a, false, b, (short)0, acc2, false, false);
    }
    if (half == 0) {                              // lanes 0-15 hold rows 0-7
        #pragma unroll
        for (int bb = 0; bb < 4; ++bb) {
            float z = acc2[bb] + acc2[bb + 4];
            s_out[bb*CCH + ncol] = 1.0f / (1.0f + expf(-z));
        }
    }
}

// ---------------- Kernel 3: scrambled 3x3 gather + pool/center/confidence ----
// Output (b',m',c').  Source plane: b = c'&3, c = b'*64 + (c'>>2).
// Patch slot k': t = k'*4096 + m'; keypoint m_i = t/9, offset k = t%9.
// Block = 256 threads (one per c'), handles 8 consecutive m' for a fixed b'.
// blockIdx: b' slow (512 blocks per b') -> 151MB plane set per phase, L2-resident.
__global__ __launch_bounds__(256)
void patch_main(const float* __restrict__ kpts, const float* __restrict__ seg,
                const float* __restrict__ conv_w, const float* __restrict__ conv_b,
                const float* __restrict__ lin_w, const float* __restrict__ lin_b,
                const float* __restrict__ s_buf, float* __restrict__ out) {
    __shared__ int   pix[36];     // [k'][b]
    __shared__ float wred[8];
    int tid   = threadIdx.x;                 // c'
    int bp    = blockIdx.x >> 9;             // b'
    int mbase = (blockIdx.x & 511) * 8;

    float w[9];
    #pragma unroll
    for (int k = 0; k < 9; ++k) w[k] = conv_w[k];
    float cb = conv_b[0];
    float lb = lin_b[0];
    float lw = lin_w[tid];
    float sval = s_buf[bp*CCH + tid];

    int bsrc = tid & 3;
    int csrc = bp*64 + (tid >> 2);
    const float* plane = seg + (size_t)(bsrc*CCH + csrc) * HW;

    for (int mi = 0; mi < 8; ++mi) {
        int mp = mbase + mi;
        __syncthreads();                      // protect pix/wred reuse
        if (tid < 36) {
            int kp = tid >> 2, bb = tid & 3;
            int t    = kp * MKPT + mp;
            int kidx = t / 9;
            int k    = t - kidx * 9;
            float kx = kpts[(bb*MKPT + kidx)*2 + 0];
            float ky = kpts[(bb*MKPT + kidx)*2 + 1];
            int x = (int)floorf(kx * 384.0f);
            int y = (int)floorf(ky * 384.0f);
            int xo = min(max(x + (k/3 - 1), 0), HDIM-1);
            int yo = min(max(y + (k%3 - 1), 0), HDIM-1);
            pix[tid] = xo * HDIM + yo;
        }
        __syncthreads();

        float pool = cb, center = 0.f;
        #pragma unroll
        for (int kp = 0; kp < 9; ++kp) {
            float v = plane[pix[kp*4 + bsrc]];
            pool = fmaf(w[kp], v, pool);
            if (kp == 4) center = v;
        }

        // confidence = sum_{c'} center*lin_w[c'] + lin_b  (block reduction)
        float p = center * lw;
        #pragma unroll
        for (int off = 16; off > 0; off >>= 1)
            p += __shfl_xor(p, off, 32);
        if ((tid & 31) == 0) wred[tid >> 5] = p;
        __syncthreads();
        float conf = lb;
        #pragma unroll
        for (int q = 0; q < 8; ++q) conf += wred[q];

        out[((size_t)(bp*MKPT + mp))*CCH + tid] = (1.0f - conf)*pool + center + sval;
    }
}

extern "C" void kernel_launch(void* const* d_in, const int* in_sizes, int n_in,
                              void* d_out, int out_size, void* d_ws, size_t ws_size,
                              hipStream_t stream) {
    const float* kpts = (const float*)d_in[0];   // (4,4096,2)
    const float* seg  = (const float*)d_in[1];   // (4,256,384,384)
    const float* cw   = (const float*)d_in[2];   // (9,)
    const float* cbp  = (const float*)d_in[3];   // (1,)
    const float* lw   = (const float*)d_in[4];   // (1,256)
    const float* lbp  = (const float*)d_in[5];   // (1,)
    const float* w1   = (const float*)d_in[6];   // (256,256)
    const float* w2   = (const float*)d_in[7];   // (256,256)
    float* out  = (float*)d_out;                 // (4,4096,256)
    float* avg  = (float*)d_ws;                  // 1024 f32
    float* mx   = avg + NPLANE;                  // 1024 f32
    float* sbuf = mx + NPLANE;                   // 1024 f32

    plane_reduce<<<NPLANE, 256, 0, stream>>>(seg, avg, mx);
    ca_mlp<<<1, 512, 0, stream>>>(avg, mx, w1, w2, sbuf);
    patch_main<<<2048, 256, 0, stream>>>(kpts, seg, cw, cbp, lw, lbp, sbuf, out);
}